// GNNLayer_69861938036806
// MI455X (gfx1250) — compile-verified
//
#include <hip/hip_runtime.h>
#include <math.h>

// Workspace layout (floats): A_pre[n_node*128] | agg[n_node*128] | R_pre[n_rel*128] | K4[4*128]
// Required ws: (2*100000 + 1005)*128*4 + 2KB ~= 103 MB.

typedef __attribute__((ext_vector_type(2))) float v2f;
typedef __attribute__((ext_vector_type(8))) float v8f;

// ---------------------------------------------------------------------------
// Zero a float buffer.
__global__ void zero_f32(float* __restrict__ p, long long n) {
    long long i = (long long)blockIdx.x * blockDim.x + threadIdx.x;
    if (i < n) p[i] = 0.0f;
}

// ---------------------------------------------------------------------------
// K4[c][j] = concat(kg_table[c>>1], kg_table[c&1]) @ Wkg_w + Wkg_b   (4 x 128)
__global__ void kg_precompute(const float* __restrict__ kg_table,
                              const float* __restrict__ Wkg_w,   // (256,128)
                              const float* __restrict__ Wkg_b,   // (128,)
                              float* __restrict__ K4) {
    int t = threadIdx.x;           // 512 threads
    int c = t >> 7;
    int j = t & 127;
    const float* kh = kg_table + (c >> 1) * 128;
    const float* kt = kg_table + (c & 1) * 128;
    float acc = Wkg_b[j];
    #pragma unroll 4
    for (int i = 0; i < 128; ++i) {
        acc += kh[i] * Wkg_w[i * 128 + j];
        acc += kt[i] * Wkg_w[(128 + i) * 128 + j];
    }
    K4[c * 128 + j] = acc;
}

// ---------------------------------------------------------------------------
// C[M x 128] = A[M x 128] @ B[128 x 128], fp32 via V_WMMA_F32_16X16X4_F32.
// Block = 256 threads = 8 waves, covering a 64-row M stripe.
//  - A stripe (64x128) staged in LDS with float4 loads (global_load_b128 +
//    ds_store_b128); row stride 132 floats kills the 16-way bank conflict on
//    fragment reads (132 % 64 = 4 -> 16 distinct banks for the 16 M-lanes).
//  - Wave w owns N-tile [16w, 16w+16); its B fragments (32 x v2f = 64 VGPRs)
//    are loaded from global ONCE per block and reused across 4 M-tiles.
//  - 128 v_wmma_f32_16x16x4_f32 per wave per block.
__global__ __launch_bounds__(256) void gemm128_wmma_f32(
        const float* __restrict__ A, const float* __restrict__ B,
        float* __restrict__ C, int M) {
    __shared__ float As[64 * 132];
    const int m0  = blockIdx.x * 64;
    const int tid = threadIdx.x;

    // Cooperative float4 staging of the 64x128 A stripe (rows clamped; clamped
    // rows are never stored to C).
    for (int i = tid; i < 64 * 32; i += 256) {
        int r  = i >> 5;       // row in stripe
        int kq = i & 31;       // float4 index in row
        int row = m0 + r;
        if (row >= M) row = M - 1;
        float4 v = ((const float4*)(A + (long long)row * 128))[kq];
        *(float4*)(&As[r * 132 + kq * 4]) = v;
    }

    const int wave = tid >> 5;
    const int lane = tid & 31;
    const int n0   = wave * 16;
    const int h    = lane >> 4;   // lane half selects the K pair
    const int mn   = lane & 15;   // A: M index, B: N index (ISA fragment layout)

    // B fragments for this wave's N-tile, held in registers for the whole block.
    v2f bf[32];
    #pragma unroll
    for (int i = 0; i < 32; ++i) {
        const float* bp = B + (i * 4 + 2 * h) * 128 + n0 + mn;
        bf[i].x = bp[0];
        bf[i].y = bp[128];
    }

    __syncthreads();

    for (int mt = 0; mt < 4; ++mt) {
        const float* arow = As + (mt * 16 + mn) * 132 + 2 * h;
        v8f acc = {0.f, 0.f, 0.f, 0.f, 0.f, 0.f, 0.f, 0.f};
        #pragma unroll
        for (int i = 0; i < 32; ++i) {
            v2f a;
            a.x = arow[i * 4];
            a.y = arow[i * 4 + 1];
            acc = __builtin_amdgcn_wmma_f32_16x16x4_f32(
                /*neg_a=*/false, a, /*neg_b=*/false, bf[i],
                /*c_mod=*/(short)0, acc, /*reuse_a=*/false, /*reuse_b=*/false);
        }
        // D layout: VGPR r -> row (r + 8*h), col = n0 + mn.
        #pragma unroll
        for (int r = 0; r < 8; ++r) {
            int row = m0 + mt * 16 + r + 8 * h;
            if (row < M) C[(long long)row * 128 + n0 + mn] = acc[r];
        }
    }
}

// ---------------------------------------------------------------------------
// Per-edge gate + message scatter. One 128-thread block per edge; thread j is
// channel j. alpha = sigmoid(relu(A_pre[sub]+R_pre[rel]+K4[c]) . w + b);
// atomicAdd alpha*(hidden[sub]+rela[rel]) into agg[obj].
__global__ __launch_bounds__(128) void edge_gate_scatter(
        const float* __restrict__ Apre, const float* __restrict__ Rpre,
        const float* __restrict__ K4,
        const float* __restrict__ hidden, const float* __restrict__ rela,
        const long long* __restrict__ edges,
        const float* __restrict__ walpha_w, const float* __restrict__ walpha_b,
        const int* __restrict__ left_num_p,
        float* __restrict__ agg, int n_edge) {
    const int e = blockIdx.x;
    if (e >= n_edge) return;
    const int j = threadIdx.x;  // 0..127

    const long long* ed = edges + (long long)e * 6;
    const int head = (int)ed[1];
    const int rel  = (int)ed[2];
    const int tail = (int)ed[3];
    const int sub  = (int)ed[4];
    const int obj  = (int)ed[5];

    const int left = left_num_p[0];   // low 32 bits (valid for i32 or LE i64)
    const int c = ((head >= left) ? 2 : 0) | ((tail >= left) ? 1 : 0);

    float pre = Apre[(long long)sub * 128 + j]
              + Rpre[rel * 128 + j]
              + K4[c * 128 + j];
    float r = fmaxf(pre, 0.0f) * walpha_w[j];

    // wave32 reduction, then combine the 4 waves via LDS
    #pragma unroll
    for (int off = 16; off > 0; off >>= 1)
        r += __shfl_xor(r, off, 32);

    __shared__ float partial[4];
    if ((j & 31) == 0) partial[j >> 5] = r;
    __syncthreads();
    const float dot = partial[0] + partial[1] + partial[2] + partial[3];

    const float alpha = 1.0f / (1.0f + __expf(-(dot + walpha_b[0])));
    const float msg = alpha * (hidden[(long long)sub * 128 + j] + rela[rel * 128 + j]);
    atomicAdd(&agg[(long long)obj * 128 + j], msg);
}

// ---------------------------------------------------------------------------
extern "C" void kernel_launch(void* const* d_in, const int* in_sizes, int n_in,
                              void* d_out, int out_size, void* d_ws, size_t ws_size,
                              hipStream_t stream) {
    const float*     hidden   = (const float*)d_in[0];
    const long long* edges    = (const long long*)d_in[1];   // int64 per reference
    const float*     kg_table = (const float*)d_in[2];
    const float*     rela     = (const float*)d_in[3];
    const float*     Ws       = (const float*)d_in[4];
    const float*     Wr       = (const float*)d_in[5];
    const float*     Wkg_w    = (const float*)d_in[6];
    const float*     Wkg_b    = (const float*)d_in[7];
    const float*     walpha_w = (const float*)d_in[8];
    const float*     walpha_b = (const float*)d_in[9];
    const float*     W_h      = (const float*)d_in[10];
    const int*       left_p   = (const int*)d_in[12];        // low word is the value

    const int n_node = in_sizes[0] / 128;
    const int n_edge = in_sizes[1] / 6;
    const int n_rel  = in_sizes[3] / 128;

    float* out  = (float*)d_out;
    float* ws   = (float*)d_ws;
    float* Apre = ws;
    float* agg  = Apre + (size_t)n_node * 128;
    float* Rpre = agg + (size_t)n_node * 128;
    float* K4   = Rpre + (size_t)n_rel * 128;

    // 1) agg = 0
    const long long tot = (long long)n_node * 128;
    zero_f32<<<(unsigned)((tot + 255) / 256), 256, 0, stream>>>(agg, tot);

    // 2) 4-entry kg attention table
    kg_precompute<<<1, 512, 0, stream>>>(kg_table, Wkg_w, Wkg_b, K4);

    // 3) A_pre = hidden @ Ws_attn ; R_pre = rela_embed @ Wr_attn  (fp32 WMMA)
    gemm128_wmma_f32<<<(n_node + 63) / 64, 256, 0, stream>>>(hidden, Ws, Apre, n_node);
    gemm128_wmma_f32<<<(n_rel  + 63) / 64, 256, 0, stream>>>(rela,   Wr, Rpre, n_rel);

    // 4) gated message scatter into agg
    edge_gate_scatter<<<n_edge, 128, 0, stream>>>(Apre, Rpre, K4, hidden, rela,
                                                  edges, walpha_w, walpha_b,
                                                  left_p, agg, n_edge);

    // 5) out = agg @ W_h  (fp32 WMMA)
    gemm128_wmma_f32<<<(n_node + 63) / 64, 256, 0, stream>>>(agg, W_h, out, n_node);
}